// Cheb_43568148250776
// MI455X (gfx1250) — compile-verified
//
#include <hip/hip_runtime.h>

#define NNODES 50000
#define EEDGES 800000
#define FDIM   128

typedef float v2f __attribute__((ext_vector_type(2)));
typedef float v8f __attribute__((ext_vector_type(8)));

static __device__ __forceinline__ int imin(int a, int b) { return a < b ? a : b; }

// ---------------- utility kernels ----------------

__global__ void zero_f32(float* p, long long n) {
    long long i4 = ((long long)blockIdx.x * blockDim.x + threadIdx.x) * 4;
    if (i4 < n) *(float4*)(p + i4) = make_float4(0.f, 0.f, 0.f, 0.f);
}

__global__ void negcopy_f32(float* o, const float* in, long long n) {
    long long i4 = ((long long)blockIdx.x * blockDim.x + threadIdx.x) * 4;
    if (i4 < n) {
        float4 v = *(const float4*)(in + i4);
        *(float4*)(o + i4) = make_float4(-v.x, -v.y, -v.z, -v.w);
    }
}

// ---------------- graph norm kernels ----------------

__global__ void degree_kernel(const int* __restrict__ src, const int* __restrict__ dst,
                              float* __restrict__ deg, int E) {
    int e = blockIdx.x * blockDim.x + threadIdx.x;
    if (e >= E) return;
    int s = src[e];
    if (s != dst[e]) atomicAdd(&deg[s], 1.0f);
}

__global__ void dinv_kernel(const float* __restrict__ deg, float* __restrict__ dinv, int n) {
    int i = blockIdx.x * blockDim.x + threadIdx.x;
    if (i >= n) return;
    float d = deg[i];
    dinv[i] = (d > 0.f) ? rsqrtf(fmaxf(d, 1.f)) : 0.f;
}

__global__ void edgew_kernel(const int* __restrict__ src, const int* __restrict__ dst,
                             const float* __restrict__ dinv, float* __restrict__ w, int E) {
    int e = blockIdx.x * blockDim.x + threadIdx.x;
    if (e >= E) return;
    int s = src[e], d = dst[e];
    w[e] = (s != d) ? (-dinv[s] * dinv[d]) : 0.f;
}

// ---------------- SpMM scatter: vout[dst] += scale*w[e]*vin[src] ----------------
// One thread per (edge, 4 features); per-scatter working set (~55MB) is
// L2-resident on MI455X (192MB L2), so gathers + f32 atomics resolve on-die.

__global__ void scatter_kernel(const int* __restrict__ src, const int* __restrict__ dst,
                               const float* __restrict__ w,
                               const float* __restrict__ vin, float* __restrict__ vout,
                               int E, float scale) {
    long long t = (long long)blockIdx.x * blockDim.x + threadIdx.x;
    int e = (int)(t >> 5);
    if (e >= E) return;
    int f = ((int)t & 31) * 4;
    float we = w[e] * scale;
    if (we == 0.0f) return;
    int s = src[e], d = dst[e];
    float4 v = *(const float4*)(vin + (size_t)s * FDIM + f);
    float* o = vout + (size_t)d * FDIM + f;
    atomicAdd(o + 0, we * v.x);
    atomicAdd(o + 1, we * v.y);
    atomicAdd(o + 2, we * v.z);
    atomicAdd(o + 3, we * v.w);
}

// ---------------- WMMA GEMM: C[M x 128] (+)= A[M x 128] * B[128 x 128] ----------------
// 8 wave32 per block; wave owns a 16-row stripe; B staged in LDS in two halves
// with a K-pair swizzle: element (k,n) -> Bs[(k>>1)*288 + n*2 + (k&1)], so each
// WMMA B-operand is one aligned ds_load_b64 (pairs merge into ds_load_2addr_b64);
// +32-float pad makes lanes 16-31 (kp+1) land on banks 32-63 -> conflict-free.
// All 8 B operands of a k-step are gathered into breg[] before the WMMA burst so
// DS latency is covered by the WMMA stream instead of a wait per operand pair.
// initMode==1: acc = bias broadcast; initMode==0: acc = C (accumulate).
// doRelu applied in the store epilogue.

__launch_bounds__(256)
__global__ void gemm_wmma(const float* __restrict__ A, const float* __restrict__ B,
                          const float* __restrict__ bias, float* __restrict__ C,
                          int M, int initMode, int doRelu) {
    __shared__ float Bs[32 * 288];   // 36 KB

    const int lane  = threadIdx.x & 31;
    const int wid   = threadIdx.x >> 5;
    const int lmod  = lane & 15;
    const int lhi   = lane >> 4;                    // 0 or 1
    const int mbase = blockIdx.x * 128 + wid * 16;

    // Clamped row: loads always in-bounds; invalid rows are never stored.
    const int mclamp = imin(mbase + lmod, M - 1);
    const float* __restrict__ Arow = A + (size_t)mclamp * FDIM;

    v8f acc[8];
    if (initMode == 1) {
#pragma unroll
        for (int nt = 0; nt < 8; ++nt) {
            float bv = bias[nt * 16 + lmod];
#pragma unroll
            for (int r = 0; r < 8; ++r) acc[nt][r] = bv;
        }
    } else {
#pragma unroll
        for (int r = 0; r < 8; ++r) {
            const float* Crow = C + (size_t)imin(mbase + r + 8 * lhi, M - 1) * FDIM;
#pragma unroll
            for (int nt = 0; nt < 8; ++nt)
                acc[nt][r] = Crow[nt * 16 + lmod];
        }
    }

    for (int half = 0; half < 2; ++half) {
        __syncthreads();
        // Stage B[half*64 .. +63][0..127] into swizzled LDS (8192 floats).
        {
            const float* Bsrc = B + (size_t)half * 64 * 128;
#pragma unroll
            for (int i = 0; i < 8; ++i) {
                int idx = (i * 256 + threadIdx.x) * 4;   // linear float index
                int k = idx >> 7;
                int n = idx & 127;
                float4 v = *(const float4*)(Bsrc + idx);
                int base = (k >> 1) * 288 + (k & 1) + n * 2;
                Bs[base + 0] = v.x;
                Bs[base + 2] = v.y;
                Bs[base + 4] = v.z;
                Bs[base + 6] = v.w;
            }
        }
        __syncthreads();

        // Preload this half's 16 A fragments (one load cluster, no branches).
        v2f areg[16];
#pragma unroll
        for (int kt = 0; kt < 16; ++kt)
            areg[kt] = *(const v2f*)(Arow + half * 64 + kt * 4 + lhi * 2);

#pragma unroll
        for (int kt = 0; kt < 16; ++kt) {
            const int kp = kt * 2 + lhi;             // K-pair row in swizzled Bs
            const float* brow = Bs + kp * 288 + lmod * 2;
            v2f breg[8];
#pragma unroll
            for (int nt = 0; nt < 8; ++nt)
                breg[nt] = *(const v2f*)(brow + nt * 32);
#pragma unroll
            for (int nt = 0; nt < 8; ++nt)
                acc[nt] = __builtin_amdgcn_wmma_f32_16x16x4_f32(
                    false, areg[kt], false, breg[nt], (short)0, acc[nt], false, false);
        }
    }

    // Store: one guard per row covering 8 column-tiles; optional fused ReLU.
#pragma unroll
    for (int r = 0; r < 8; ++r) {
        int m = mbase + r + 8 * lhi;
        if (m < M) {
            float* Crow = C + (size_t)m * FDIM;
#pragma unroll
            for (int nt = 0; nt < 8; ++nt) {
                float v = acc[nt][r];
                if (doRelu) v = fmaxf(v, 0.f);
                Crow[nt * 16 + lmod] = v;
            }
        }
    }
}

// ---------------- final linear: out[n] = dot(H[n,:], Wl) + bl ----------------

__global__ void linear_out(const float* __restrict__ H, const float* __restrict__ Wl,
                           const float* __restrict__ bl, float* __restrict__ out, int M) {
    int lane = threadIdx.x & 31;
    int wid  = threadIdx.x >> 5;
    int node = blockIdx.x * 8 + wid;
    if (node >= M) return;
    float4 h4 = *(const float4*)(H + (size_t)node * FDIM + lane * 4);
    float4 w4 = *(const float4*)(Wl + lane * 4);
    float s = h4.x * w4.x + h4.y * w4.y + h4.z * w4.z + h4.w * w4.w;
#pragma unroll
    for (int off = 16; off > 0; off >>= 1)
        s += __shfl_down(s, off, 32);
    if (lane == 0) out[node] = s + bl[0];
}

// ---------------- host orchestration ----------------

extern "C" void kernel_launch(void* const* d_in, const int* in_sizes, int n_in,
                              void* d_out, int out_size, void* d_ws, size_t ws_size,
                              hipStream_t stream) {
    (void)in_sizes; (void)n_in; (void)out_size; (void)ws_size;

    const float* x  = (const float*)d_in[0];
    const int*   ei = (const int*)d_in[1];
    const float* W0 = (const float*)d_in[2];
    const float* b0 = (const float*)d_in[3];
    const float* W1 = (const float*)d_in[4];
    const float* b1 = (const float*)d_in[5];
    const float* W2 = (const float*)d_in[6];
    const float* b2 = (const float*)d_in[7];
    const float* Wl = (const float*)d_in[8];
    const float* bl = (const float*)d_in[9];
    float* out = (float*)d_out;

    const int Nn = NNODES, Ee = EEDGES;
    const long long NF = (long long)Nn * FDIM;

    char* p = (char*)d_ws;
    auto carve = [&](size_t bytes) -> float* {
        char* r = p;
        p += (bytes + 255) & ~(size_t)255;
        return (float*)r;
    };
    float* deg  = carve((size_t)Nn * 4);
    float* dinv = carve((size_t)Nn * 4);
    float* wE   = carve((size_t)Ee * 4);
    float* Ta   = carve((size_t)NF * 4);
    float* Tb   = carve((size_t)NF * 4);
    float* Tc   = carve((size_t)NF * 4);
    float* Oa   = carve((size_t)NF * 4);
    float* Ob   = carve((size_t)NF * 4);

    const int* srcI = ei;
    const int* dstI = ei + Ee;

    const unsigned gN4  = (unsigned)((NF / 4 + 255) / 256);
    const unsigned gE   = (unsigned)((Ee + 255) / 256);
    const unsigned gNn  = (unsigned)((Nn + 255) / 256);
    const unsigned gSc  = (unsigned)(((long long)Ee * 32 + 255) / 256);
    const unsigned gGmm = (unsigned)((Nn + 127) / 128);

    // ---- normalization ----
    zero_f32<<<(unsigned)((Nn / 4 + 255) / 256), 256, 0, stream>>>(deg, Nn);
    degree_kernel<<<gE, 256, 0, stream>>>(srcI, dstI, deg, Ee);
    dinv_kernel<<<gNn, 256, 0, stream>>>(deg, dinv, Nn);
    edgew_kernel<<<gE, 256, 0, stream>>>(srcI, dstI, dinv, wE, Ee);

    auto spmm = [&](const float* vin, float* vout, float scale) {
        scatter_kernel<<<gSc, 256, 0, stream>>>(srcI, dstI, wE, vin, vout, Ee, scale);
    };
    auto gemm = [&](const float* A, const float* Bw, const float* bias, float* C,
                    int init, int relu) {
        gemm_wmma<<<gGmm, 256, 0, stream>>>(A, Bw, bias, C, Nn, init, relu);
    };

    auto layer = [&](const float* X, const float* Wk, const float* bias, float* O, bool doRelu) {
        const int R = doRelu ? 1 : 0;
        // k = 0 : Tx0 = X
        gemm(X, Wk, bias, O, 1, 0);
        // k = 1 : Tx1 = L X   (in Ta)
        zero_f32<<<gN4, 256, 0, stream>>>(Ta, NF);
        spmm(X, Ta, 1.0f);
        gemm(Ta, Wk + 1 * 16384, bias, O, 0, 0);
        // k = 2 : Tx2 = 2 L Tx1 - Tx0   (in Tb)
        negcopy_f32<<<gN4, 256, 0, stream>>>(Tb, X, NF);
        spmm(Ta, Tb, 2.0f);
        gemm(Tb, Wk + 2 * 16384, bias, O, 0, 0);
        // k = 3 : Tx3 = 2 L Tx2 - Tx1   (in Tc)
        negcopy_f32<<<gN4, 256, 0, stream>>>(Tc, Ta, NF);
        spmm(Tb, Tc, 2.0f);
        gemm(Tc, Wk + 3 * 16384, bias, O, 0, 0);
        // k = 4 : Tx4 = 2 L Tx3 - Tx2   (reuse Ta) ; fused ReLU on final accumulate
        negcopy_f32<<<gN4, 256, 0, stream>>>(Ta, Tb, NF);
        spmm(Tc, Ta, 2.0f);
        gemm(Ta, Wk + 4 * 16384, bias, O, 0, R);
    };

    layer(x,  W0, b0, Oa, true);
    layer(Oa, W1, b1, Ob, true);
    layer(Ob, W2, b2, Oa, false);

    linear_out<<<(unsigned)((Nn + 7) / 8), 256, 0, stream>>>(Oa, Wl, bl, out, Nn);
}